// ExpMatch_11476152615412
// MI455X (gfx1250) — compile-verified
//
#include <hip/hip_runtime.h>
#include <hip/hip_bf16.h>

typedef float v2f __attribute__((ext_vector_type(2)));
typedef float v8f __attribute__((ext_vector_type(8)));

#define B_ 256
#define P_ 64
#define L_ 16
#define D_ 64
#define IMGF_ 512

// ---------------------------------------------------------------------------
// Kernel 1: gathered projection  out[r,:] = l2norm(img[id[r]] @ W^T + b)
// 512 rows total (256 qry + 256 res). One block = 16 rows, 4 waves, each wave
// owns a 16-wide N tile and runs K=512 in steps of 4 via V_WMMA_F32_16X16X4_F32.
// ---------------------------------------------------------------------------
__global__ __launch_bounds__(128) void proj_kernel(
    const int* __restrict__ qry_id, const int* __restrict__ res_id,
    const float* __restrict__ img, const float* __restrict__ W,
    const float* __restrict__ bias,
    float* __restrict__ qry_fea, float* __restrict__ res_fea)
{
    __shared__ float tileX[16][516];   // +4 pad: distinct banks, 8B aligned rows
    __shared__ float tileO[16][64];
    __shared__ float rinv[16];

    const int tid  = threadIdx.x;      // 0..127
    const int wave = tid >> 5;         // 0..3  -> N tile
    const int lane = tid & 31;
    const int rowBase = blockIdx.x * 16;

    // ---- stage 16 gathered rows of X (16 x 512 f32) into LDS -------------
    {
        int r   = tid >> 3;            // 0..15
        int sub = tid & 7;             // 0..7, each loads 64 floats
        int grow = rowBase + r;
        int id   = (grow < B_) ? qry_id[grow] : res_id[grow - B_];
        const float4* src = (const float4*)(img + (long)id * IMGF_ + sub * 64);
        float4* dst = (float4*)(&tileX[r][sub * 64]);
#pragma unroll
        for (int i = 0; i < 16; ++i) dst[i] = src[i];
    }
    __syncthreads();

    // ---- WMMA K loop ------------------------------------------------------
    const int khalf = lane >> 4;       // 0: K 0,1   1: K 2,3
    const int lm    = lane & 15;       // A row / B col within tile
    const float* Wrow = W + (long)(wave * 16 + lm) * IMGF_;  // B[k][n] = W[n][k]

    v8f acc = {};
#pragma unroll 8
    for (int k0 = 0; k0 < IMGF_; k0 += 4) {
        v2f a, b;
        a.x = tileX[lm][k0 + 2 * khalf];
        a.y = tileX[lm][k0 + 2 * khalf + 1];
        b.x = Wrow[k0 + 2 * khalf];
        b.y = Wrow[k0 + 2 * khalf + 1];
        acc = __builtin_amdgcn_wmma_f32_16x16x4_f32(
            /*neg_a=*/false, a, /*neg_b=*/false, b,
            /*c_mod=*/(short)0, acc, /*reuse_a=*/false, /*reuse_b=*/false);
    }

    // ---- epilogue: bias, row l2norm, store --------------------------------
    {
        int col = wave * 16 + lm;
        float bv = bias[col];
#pragma unroll
        for (int v = 0; v < 8; ++v) {
            int row = v + 8 * khalf;   // C/D layout: VGPR v, lanes>=16 -> M=v+8
            tileO[row][col] = acc[v] + bv;
        }
    }
    __syncthreads();
    if (tid < 16) {
        float s = 0.f;
#pragma unroll
        for (int c = 0; c < 64; ++c) { float x = tileO[tid][c]; s += x * x; }
        rinv[tid] = 1.0f / fmaxf(sqrtf(s), 1e-12f);
    }
    __syncthreads();
    {
        int r  = tid >> 3;
        int c0 = (tid & 7) * 8;
        int grow = rowBase + r;
        float* dst = (grow < B_) ? (qry_fea + (long)grow * D_)
                                 : (res_fea + (long)(grow - B_) * D_);
        float inv = rinv[r];
#pragma unroll
        for (int i = 0; i < 8; ++i) dst[c0 + i] = tileO[r][c0 + i] * inv;
    }
}

// ---------------------------------------------------------------------------
// Kernel 2: per-(b,p) path embedding processing.
// 64 threads (thread = D column). Produces path_res[b,p,:] and the
// p-dependent attention logit partial (softmax is shift-invariant in p, so the
// ui_mul / bias terms are dropped exactly).
// ---------------------------------------------------------------------------
__global__ __launch_bounds__(64) void path_kernel(
    const int* __restrict__ path, const int* __restrict__ mask,
    const float* __restrict__ meta,
    const float* __restrict__ qry_fea, const float* __restrict__ res_fea,
    const float* __restrict__ h_att_w,
    float* __restrict__ path_res, float* __restrict__ logits)
{
    const int d  = threadIdx.x;        // 0..63
    const int bp = blockIdx.x;         // b*P + p
    const int b  = bp >> 6;

    __shared__ float red[16][64];
    __shared__ float rn[16];

    const int* pp = path + (long)bp * L_;
    const int* mm = mask + (long)bp * L_;

    float val[16];
#pragma unroll
    for (int l = 0; l < L_; ++l) {
        int idx = pp[l];
        val[l] = meta[(long)idx * D_ + d];
        red[l][d] = val[l] * val[l];
    }
    __syncthreads();
    if (d < 16) {
        float s = 0.f;
#pragma unroll
        for (int c = 0; c < 64; ++c) s += red[d][c];
        rn[d] = 1.0f / fmaxf(sqrtf(s), 1e-12f);
    }
    __syncthreads();

    float pm[8];
#pragma unroll
    for (int j = 0; j < 8; ++j) {
        float p0 = val[2*j]     * rn[2*j]     * (float)mm[2*j];
        float p1 = val[2*j + 1] * rn[2*j + 1] * (float)mm[2*j + 1];
        pm[j] = p0 + p1;
    }

    float i2 = 0.f, i3 = 0.f, i4 = 0.f;
#pragma unroll
    for (int k = 0; k < 7; ++k) i2 += pm[k] * pm[k+1];
#pragma unroll
    for (int k = 0; k < 6; ++k) i3 += pm[k] * pm[k+1] * pm[k+2];
#pragma unroll
    for (int k = 0; k < 5; ++k) i4 += pm[k] * pm[k+1] * pm[k+2] * pm[k+3];
    i2 *= (1.0f / 7.0f); i3 *= (1.0f / 6.0f); i4 *= (1.0f / 5.0f);

    red[0][d] = i2 * i2; red[1][d] = i3 * i3; red[2][d] = i4 * i4;
    __syncthreads();
    if (d < 3) {
        float s = 0.f;
#pragma unroll
        for (int c = 0; c < 64; ++c) s += red[d][c];
        rn[d] = 1.0f / fmaxf(sqrtf(s), 1e-12f);
    }
    __syncthreads();

    float pr = (i2 * rn[0] + i3 * rn[1] + i4 * rn[2]) * (1.0f / 3.0f);
    path_res[(long)bp * D_ + d] = pr;

    float q = qry_fea[b * D_ + d], r = res_fea[b * D_ + d];
    red[0][d] = -(q - r) * pr * h_att_w[d];
    __syncthreads();
    if (d == 0) {
        float s = 0.f;
#pragma unroll
        for (int c = 0; c < 64; ++c) s += red[0][c];
        logits[bp] = s;
    }
}

// ---------------------------------------------------------------------------
// Kernel 3: per-b softmax over P, pooled sum, prediction heads.
// ---------------------------------------------------------------------------
__global__ __launch_bounds__(64) void pool_kernel(
    const float* __restrict__ qry_fea, const float* __restrict__ res_fea,
    const float* __restrict__ path_res, const float* __restrict__ logits,
    const float* __restrict__ p1w, const float* __restrict__ p1b,
    const float* __restrict__ p2w, const float* __restrict__ p2b,
    float* __restrict__ out)
{
    const int t = threadIdx.x;         // 0..63 (= p for softmax, = d for pool)
    const int b = blockIdx.x;

    __shared__ float sh[64];
    __shared__ float w[64];
    __shared__ float mred[2];

    float lg = logits[b * P_ + t] * 5.0f;     // /0.2
    sh[t] = lg;
    __syncthreads();
    if (t == 0) {
        float m = sh[0];
        for (int i = 1; i < 64; ++i) m = fmaxf(m, sh[i]);
        mred[0] = m;
    }
    __syncthreads();
    float e = expf(lg - mred[0]);
    sh[t] = e;
    __syncthreads();
    if (t == 0) {
        float s = 0.f;
        for (int i = 0; i < 64; ++i) s += sh[i];
        mred[1] = s;
    }
    __syncthreads();
    w[t] = e / mred[1];
    __syncthreads();

    // pooled[d] = sum_p path_res[b,p,d] * w[p]
    float pooled = 0.f;
    const float* prb = path_res + (long)b * P_ * D_;
    for (int p = 0; p < P_; ++p) pooled += prb[p * D_ + t] * w[p];

    float q = qry_fea[b * D_ + t], r = res_fea[b * D_ + t];
    float s1p = q * r * p1w[t];
    float s2p = (r - q) * pooled * p2w[t];

    sh[t] = s1p;
    __syncthreads();              // also fences the w[] reads above
    w[t] = s2p;
    __syncthreads();
    if (t == 0) {
        float a = 0.f, c = 0.f;
        for (int i = 0; i < 64; ++i) { a += sh[i]; c += w[i]; }
        out[b] = (a + p1b[0]) + 5.0f * (c + p2b[0]);
    }
}

// ---------------------------------------------------------------------------
extern "C" void kernel_launch(void* const* d_in, const int* in_sizes, int n_in,
                              void* d_out, int out_size, void* d_ws, size_t ws_size,
                              hipStream_t stream) {
    const int*   qry_id = (const int*)d_in[0];
    const int*   res_id = (const int*)d_in[1];
    const int*   path   = (const int*)d_in[2];
    const int*   mask   = (const int*)d_in[3];
    const float* img    = (const float*)d_in[4];
    const float* Ww     = (const float*)d_in[5];
    const float* Wb     = (const float*)d_in[6];
    const float* meta   = (const float*)d_in[7];
    const float* hw     = (const float*)d_in[8];
    // d_in[9] = h_att_b: constant per-b shift, cancelled by softmax (exact)
    const float* p1w    = (const float*)d_in[10];
    const float* p1b    = (const float*)d_in[11];
    const float* p2w    = (const float*)d_in[12];
    const float* p2b    = (const float*)d_in[13];

    float* ws       = (float*)d_ws;
    float* qry_fea  = ws;                       // 256*64
    float* res_fea  = qry_fea + B_ * D_;        // 256*64
    float* path_res = res_fea + B_ * D_;        // 256*64*64
    float* logits   = path_res + (long)B_ * P_ * D_;  // 256*64

    proj_kernel<<<(2 * B_) / 16, 128, 0, stream>>>(qry_id, res_id, img, Ww, Wb,
                                                   qry_fea, res_fea);
    path_kernel<<<B_ * P_, 64, 0, stream>>>(path, mask, meta, qry_fea, res_fea,
                                            hw, path_res, logits);
    pool_kernel<<<B_, 64, 0, stream>>>(qry_fea, res_fea, path_res, logits,
                                       p1w, p1b, p2w, p2b, (float*)d_out);
}